// MultiplexAttention_56573309224530
// MI455X (gfx1250) — compile-verified
//
#include <hip/hip_runtime.h>
#include <hip/hip_bf16.h>

// Problem constants (from reference): V=3, A=4, N=50000, C=128
#define V_  3
#define A_  4
#define N_  50000
#define C_  128
#define VA_ (V_ * A_)

typedef __attribute__((ext_vector_type(2))) float v2f;
typedef __attribute__((ext_vector_type(8))) float v8f;

// ---------------------------------------------------------------------------
// Kernel 1: column sums over N for each (v,a):  s_sum[va][c] = sum_n x[v,a,n,c]
// Grid: (96, 12). Block: 256 threads = 8 waves; each thread owns 4 columns
// (float4), rows strided across the grid. Block-reduce in LDS, then atomicAdd
// the 128 per-column partials into ws.
// ---------------------------------------------------------------------------
__global__ void k_colsum(const float* __restrict__ x, float* __restrict__ s_sum) {
    const int va       = blockIdx.y;           // 0..11  (= v*4 + a)
    const int tid      = threadIdx.x;
    const int lane     = tid & 31;
    const int rowInBlk = tid >> 5;             // 0..7
    const int rowsPerIter = gridDim.x * 8;     // 768

    const float* xa = x + (size_t)va * N_ * C_;

    float4 acc = make_float4(0.f, 0.f, 0.f, 0.f);
    for (int n = blockIdx.x * 8 + rowInBlk; n < N_; n += rowsPerIter) {
        const float4 xv = *(const float4*)(xa + (size_t)n * C_ + lane * 4);
        acc.x += xv.x; acc.y += xv.y; acc.z += xv.z; acc.w += xv.w;
    }

    __shared__ float4 red[256];
    red[tid] = acc;
    __syncthreads();
    for (int s = 128; s >= 32; s >>= 1) {
        if (tid < s) {
            float4 o = red[tid + s];
            red[tid].x += o.x; red[tid].y += o.y;
            red[tid].z += o.z; red[tid].w += o.w;
        }
        __syncthreads();
    }
    if (tid < 32) {
        const float4 r = red[tid];
        float* dst = s_sum + va * C_ + lane * 4;
        atomicAdd(dst + 0, r.x);
        atomicAdd(dst + 1, r.y);
        atomicAdd(dst + 2, r.z);
        atomicAdd(dst + 3, r.w);
    }
}

// ---------------------------------------------------------------------------
// Kernel 2: T[v] = S_v(16x128, rows 4..15 zero) @ W[v](128x128) with
// V_WMMA_F32_16X16X4_F32, one wave per view; wave 3 computes beta[v,a]=s.b.
//   A 16x4 f32 layout : lanes 0-15 hold M=lane, VGPR0={K0 | K0+2}, VGPR1={K0+1 | K0+3}
//   D 16x16 f32 layout: VGPR r -> M=r (lanes 0-15) / M=r+8 (lanes 16-31)
// ---------------------------------------------------------------------------
__global__ void k_prepare(const float* __restrict__ W,
                          const float* __restrict__ bvec,
                          const float* __restrict__ s_sum,
                          float* __restrict__ t_out,
                          float* __restrict__ beta) {
    const int tid  = threadIdx.x;     // 128 threads = 4 waves
    const int wave = tid >> 5;
    const int lane = tid & 31;
    const float invN = 1.0f / (float)N_;

    if (wave < V_) {
        const int v = wave;
        const float* Wv = W + (size_t)v * C_ * C_;
        const int row   = lane & 15;             // M index this lane supplies
        const int khalf = (lane >> 4) * 2;       // 0 for lanes 0-15, 2 for 16-31
        const int srow  = (row < A_) ? row : 0;  // clamp (avoid OOB read)
        const float amask = (row < A_) ? invN : 0.0f;  // zero-pad rows 4..15

        for (int jt = 0; jt < 8; ++jt) {
            const int col = jt * 16 + (lane & 15);
            v8f acc = {0.f, 0.f, 0.f, 0.f, 0.f, 0.f, 0.f, 0.f};
#pragma unroll 4
            for (int k0 = 0; k0 < C_; k0 += 4) {
                const int ka = k0 + khalf;
                v2f afrag;
                afrag.x = s_sum[(v * A_ + srow) * C_ + ka]     * amask;
                afrag.y = s_sum[(v * A_ + srow) * C_ + ka + 1] * amask;
                v2f bfrag;                        // B[k][col] = W[v][k][col]
                bfrag.x = Wv[(size_t)ka * C_ + col];
                bfrag.y = Wv[(size_t)(ka + 1) * C_ + col];
                acc = __builtin_amdgcn_wmma_f32_16x16x4_f32(
                    false, afrag, false, bfrag, (short)0, acc, false, false);
            }
            if (lane < 16) {                      // rows M=0..3 live in VGPRs 0..3
#pragma unroll
                for (int r = 0; r < A_; ++r)
                    t_out[(v * A_ + r) * C_ + col] = acc[r];
            }
        }
    } else {
        // wave 3: beta[va] = (s_sum[va,:]/N) . b[v,:]
        if (lane < VA_) {
            const int v = lane >> 2;
            float sdot = 0.f;
            for (int d = 0; d < C_; ++d)
                sdot += s_sum[lane * C_ + d] * invN * bvec[v * C_ + d];
            beta[lane] = sdot;
        }
    }
}

// ---------------------------------------------------------------------------
// Kernel 3: fused scores + tanh + softmax(over v) + weighted combine.
// One wave32 per node; lane owns 4 contiguous columns (float4 => 512B/row/wave).
// ---------------------------------------------------------------------------
__global__ void k_fuse(const float* __restrict__ x,
                       const float* __restrict__ t,
                       const float* __restrict__ beta,
                       float* __restrict__ out) {
    const int tid  = threadIdx.x;
    const int lane = tid & 31;
    const int wid  = tid >> 5;
    const long g = (long)blockIdx.x * 8 + wid;   // node index in [0, A*N)
    const int a = (int)(g / N_);
    const int n = (int)(g % N_);

    float4 xr[V_];
    float  part[V_];
#pragma unroll
    for (int v = 0; v < V_; ++v) {
        const size_t base = ((size_t)(v * A_ + a) * N_ + n) * C_;
        xr[v] = *(const float4*)(x + base + lane * 4);
        const float4 tv = *(const float4*)(t + (v * A_ + a) * C_ + lane * 4);
        part[v] = xr[v].x * tv.x + xr[v].y * tv.y + xr[v].z * tv.z + xr[v].w * tv.w;
    }
    // wave32 reduction of the three dot products
#pragma unroll
    for (int m = 16; m >= 1; m >>= 1) {
        part[0] += __shfl_xor(part[0], m, 32);
        part[1] += __shfl_xor(part[1], m, 32);
        part[2] += __shfl_xor(part[2], m, 32);
    }
    float sc[V_];
#pragma unroll
    for (int v = 0; v < V_; ++v)
        sc[v] = tanhf(part[v] + beta[v * A_ + a]);

    const float mx = fmaxf(sc[0], fmaxf(sc[1], sc[2]));
    const float e0 = expf(sc[0] - mx);
    const float e1 = expf(sc[1] - mx);
    const float e2 = expf(sc[2] - mx);
    const float inv = 1.0f / (e0 + e1 + e2);
    const float w0 = e0 * inv, w1 = e1 * inv, w2 = e2 * inv;

    float4 o;
    o.x = w0 * xr[0].x + w1 * xr[1].x + w2 * xr[2].x;
    o.y = w0 * xr[0].y + w1 * xr[1].y + w2 * xr[2].y;
    o.z = w0 * xr[0].z + w1 * xr[1].z + w2 * xr[2].z;
    o.w = w0 * xr[0].w + w1 * xr[1].w + w2 * xr[2].w;
    *(float4*)(out + ((size_t)a * N_ + n) * C_ + lane * 4) = o;
}

// ---------------------------------------------------------------------------
// Launch: memset s_sum -> colsum -> WMMA prepare -> fused combine.
// ws layout (floats): [0,1536) s_sum | [1536,3072) t | [3072,3084) beta
// ---------------------------------------------------------------------------
extern "C" void kernel_launch(void* const* d_in, const int* in_sizes, int n_in,
                              void* d_out, int out_size, void* d_ws, size_t ws_size,
                              hipStream_t stream) {
    const float* x = (const float*)d_in[0];
    const float* W = (const float*)d_in[1];
    const float* b = (const float*)d_in[2];
    float* out = (float*)d_out;

    float* s_sum = (float*)d_ws;
    float* t     = s_sum + VA_ * C_;
    float* beta  = t + VA_ * C_;

    hipMemsetAsync(s_sum, 0, VA_ * C_ * sizeof(float), stream);

    dim3 g1(96, VA_);
    k_colsum<<<g1, 256, 0, stream>>>(x, s_sum);
    k_prepare<<<1, 128, 0, stream>>>(W, b, s_sum, t, beta);
    k_fuse<<<(A_ * N_) / 8, 256, 0, stream>>>(x, t, beta, out);
}